// Model_29360396436014
// MI455X (gfx1250) — compile-verified
//
#include <hip/hip_runtime.h>
#include <stdint.h>

#define B_N     8
#define A_N     19206
#define C_N     91
#define CC_N    90        // classes excluding background
#define K_N     256       // per-class candidates
#define MAXDET  100
#define IOU_THR 0.6f

// ---------------------------------------------------------------------------
// CDNA5 async global->LDS helpers (builtin confirmed present; params are
// typed int pointers in AS1/AS3 per the round-1 diagnostic)
// ---------------------------------------------------------------------------
#if defined(__HIP_DEVICE_COMPILE__)
typedef __attribute__((address_space(1))) int as1_int;
typedef __attribute__((address_space(3))) int as3_int;
#endif

__device__ __forceinline__ void async_ld_f32_to_lds(const float* g, float* l) {
#if defined(__HIP_DEVICE_COMPILE__)
#if __has_builtin(__builtin_amdgcn_global_load_async_to_lds_b32)
    // (global src, lds dst, imm offset, cpol). Generic->AS round-trips via
    // integer are exact on gfx1250 (global aperture identity; LDS = addr[31:0]).
    __builtin_amdgcn_global_load_async_to_lds_b32(
        (as1_int*)(uintptr_t)g, (as3_int*)(uintptr_t)l, 0, 0);
#else
    asm volatile("global_load_async_to_lds_b32 %0, %1, off"
                 :: "v"((unsigned)(uintptr_t)l),
                    "v"((unsigned long long)(uintptr_t)g)
                 : "memory");
#endif
#else
    (void)g; (void)l;
#endif
}

__device__ __forceinline__ void wait_async_zero() {
#if defined(__HIP_DEVICE_COMPILE__)
#if __has_builtin(__builtin_amdgcn_s_wait_asynccnt)
    __builtin_amdgcn_s_wait_asynccnt(0);
#else
    asm volatile("s_wait_asynccnt 0" ::: "memory");
#endif
#endif
}

// ---------------------------------------------------------------------------
// Kernel 1: decode boxes.  dec[b][a][4] = (y1,x1,y2,x2)
// ---------------------------------------------------------------------------
__global__ void __launch_bounds__(256)
decode_kernel(const float* __restrict__ enc, const float* __restrict__ anchors,
              float* __restrict__ dec) {
    int i = blockIdx.x * blockDim.x + threadIdx.x;      // over B*A
    if (i >= B_N * A_N) return;
    int a = i % A_N;
#if defined(__HIP_DEVICE_COMPILE__)
    __builtin_prefetch(enc + (size_t)(i + 4096) * 4, 0, 0);   // global_prefetch_b8
#endif
    const float* e  = enc + (size_t)i * 4;
    const float* an = anchors + (size_t)a * 4;
    float ty = e[0] / 10.0f, tx = e[1] / 10.0f;
    float th = e[2] / 5.0f,  tw = e[3] / 5.0f;
    float ya = an[0], xa = an[1], ha = an[2], wa = an[3];
    float yc = ty * ha + ya;
    float xc = tx * wa + xa;
    float h  = expf(th) * ha;
    float w  = expf(tw) * wa;
    float* o = dec + (size_t)i * 4;
    o[0] = yc - h * 0.5f;
    o[1] = xc - w * 0.5f;
    o[2] = yc + h * 0.5f;
    o[3] = xc + w * 0.5f;
}

// ---------------------------------------------------------------------------
// Kernel 2: one workgroup per (b, class).  Stage sigmoid scores in LDS,
// radix-bisect top-256, then 100 rounds of argmax + IoU suppression.
// Dynamic LDS: A_N floats (76 KB) — legal on CDNA5 (320 KB LDS / WGP).
// ---------------------------------------------------------------------------
__global__ void __launch_bounds__(256)
topk_nms_kernel(const float* __restrict__ logits, const float* __restrict__ dec,
                float* __restrict__ nmsBox, float* __restrict__ nmsSc) {
    extern __shared__ float s_sc[];                 // A_N floats
    __shared__ float red_v[256];
    __shared__ int   red_i[256];
    __shared__ float cand_s[K_N];
    __shared__ int   cand_i[K_N];
    __shared__ int   s_cnt, s_n, s_win;
    __shared__ float s_wval;
    __shared__ float s_wbox[4];

    const int wg = blockIdx.x;                      // 0 .. B*CC-1
    const int b  = wg / CC_N;
    const int ci = wg % CC_N;                       // class value = ci + 1
    const int t  = threadIdx.x;

    // ---- phase 1: async strided gather of the class column into LDS ----
    const float* col = logits + (size_t)b * A_N * C_N + (ci + 1);
    for (int a = t; a < A_N; a += 256)
        async_ld_f32_to_lds(col + (size_t)a * C_N, &s_sc[a]);
    wait_async_zero();
    __syncthreads();

    // ---- phase 2: sigmoid in place ----
    for (int a = t; a < A_N; a += 256) {
        float x = s_sc[a];
        s_sc[a] = 1.0f / (1.0f + expf(-x));
    }
    __syncthreads();

    // ---- phase 3: bisection on float bits for the 256th-largest score ----
    // scores are in (0,1]; positive-float bit pattern is order-preserving.
    unsigned lo = 0u, hi = 0x3f800000u;             // (0, 1.0]
    for (int it = 0; it < 32 && (hi - lo) > 1u; ++it) {
        unsigned mid = lo + ((hi - lo) >> 1);
        if (t == 0) s_cnt = 0;
        __syncthreads();
        int c = 0;
        for (int a = t; a < A_N; a += 256)
            if (__float_as_uint(s_sc[a]) > mid) ++c;
        atomicAdd(&s_cnt, c);
        __syncthreads();
        int cnt = s_cnt;
        __syncthreads();
        if (cnt >= K_N) lo = mid; else hi = mid;
    }
    const unsigned T = hi;   // count(bits > T) < 256 <= count(bits >= T)

    // ---- phase 4: build candidate list (strict > T, then ties in index order)
    if (t == 0) s_n = 0;
    __syncthreads();
    for (int a = t; a < A_N; a += 256) {
        float v = s_sc[a];
        if (__float_as_uint(v) > T) {
            int p = atomicAdd(&s_n, 1);
            cand_s[p] = v; cand_i[p] = a;
        }
    }
    __syncthreads();
    if (t == 0) {
        int n = s_n;
        for (int a = 0; a < A_N && n < K_N; ++a)
            if (__float_as_uint(s_sc[a]) == T) { cand_s[n] = s_sc[a]; cand_i[n] = a; ++n; }
        for (; n < K_N; ++n) { cand_s[n] = -1.0f; cand_i[n] = 0; }   // never hit (A >> K)
    }
    __syncthreads();

    // ---- phase 5: NMS over 256 candidates, one per thread ----
    float ms = cand_s[t];
    const float* bx = dec + ((size_t)b * A_N + cand_i[t]) * 4;
    float by1 = bx[0], bx1 = bx[1], by2 = bx[2], bx2 = bx[3];
    float marea = (by2 - by1) * (bx2 - bx1);

    float* outB = nmsBox + (size_t)wg * MAXDET * 4;
    float* outS = nmsSc  + (size_t)wg * MAXDET;

    for (int d = 0; d < MAXDET; ++d) {
        // block argmax, tie-break = smaller candidate index (deterministic)
        red_v[t] = ms; red_i[t] = t;
        __syncthreads();
        for (int off = 128; off > 0; off >>= 1) {
            if (t < off) {
                float v2 = red_v[t + off]; int i2 = red_i[t + off];
                if (v2 > red_v[t] || (v2 == red_v[t] && i2 < red_i[t])) {
                    red_v[t] = v2; red_i[t] = i2;
                }
            }
            __syncthreads();
        }
        if (t == 0) { s_win = red_i[0]; s_wval = red_v[0]; }
        __syncthreads();
        const int   win  = s_win;
        const float wval = s_wval;
        if (t == win) { s_wbox[0] = by1; s_wbox[1] = bx1; s_wbox[2] = by2; s_wbox[3] = bx2; }
        __syncthreads();
        float wy1 = s_wbox[0], wx1 = s_wbox[1], wy2 = s_wbox[2], wx2 = s_wbox[3];
        bool keep = wval > 0.0f;
        if (t == 0) {
            float m = keep ? 1.0f : 0.0f;
            outB[d * 4 + 0] = wy1 * m; outB[d * 4 + 1] = wx1 * m;
            outB[d * 4 + 2] = wy2 * m; outB[d * 4 + 3] = wx2 * m;
            outS[d] = keep ? wval : 0.0f;
        }
        // suppress (applied unconditionally, exactly like the reference)
        float warea = (wy2 - wy1) * (wx2 - wx1);
        float ty1 = fmaxf(by1, wy1), tx1 = fmaxf(bx1, wx1);
        float ty2 = fminf(by2, wy2), tx2 = fminf(bx2, wx2);
        float ih = fmaxf(ty2 - ty1, 0.0f), iw = fmaxf(tx2 - tx1, 0.0f);
        float inter = ih * iw;
        float iou = inter / fmaxf(marea + warea - inter, 1e-8f);
        if (iou >= IOU_THR) ms = -1.0f;
        __syncthreads();
    }
}

// ---------------------------------------------------------------------------
// Kernel 3: per-batch final top-100 over the 9000-entry pool.
// ---------------------------------------------------------------------------
__global__ void __launch_bounds__(256)
final_topk_kernel(const float* __restrict__ nmsBox, const float* __restrict__ nmsSc,
                  float* __restrict__ out) {
    __shared__ float s_sc[CC_N * MAXDET];           // 9000 floats = 36 KB
    __shared__ float red_v[256];
    __shared__ int   red_i[256];
    __shared__ int   s_win;
    __shared__ float s_wval;

    const int b = blockIdx.x;
    const int t = threadIdx.x;
    const int N = CC_N * MAXDET;

    const float* src = nmsSc + (size_t)b * N;
    for (int i = t; i < N; i += 256) s_sc[i] = src[i];
    __syncthreads();

    for (int d = 0; d < MAXDET; ++d) {
        float best = -1e30f; int bi = 0x7fffffff;
        for (int i = t; i < N; i += 256) {
            float v = s_sc[i];
            if (v > best) { best = v; bi = i; }     // strict > keeps smallest idx
        }
        red_v[t] = best; red_i[t] = bi;
        __syncthreads();
        for (int off = 128; off > 0; off >>= 1) {
            if (t < off) {
                float v2 = red_v[t + off]; int i2 = red_i[t + off];
                if (v2 > red_v[t] || (v2 == red_v[t] && i2 < red_i[t])) {
                    red_v[t] = v2; red_i[t] = i2;
                }
            }
            __syncthreads();
        }
        if (t == 0) { s_win = red_i[0]; s_wval = red_v[0]; }
        __syncthreads();
        const int win = s_win;
        if (t == 0) {
            const float* bb = nmsBox + ((size_t)b * N + win) * 4;
            float* o = out + ((size_t)b * MAXDET + d) * 6;
            o[0] = bb[0]; o[1] = bb[1]; o[2] = bb[2]; o[3] = bb[3];
            o[4] = (float)(1 + win / MAXDET);       // class value
            o[5] = s_wval;
            s_sc[win] = -1e30f;                     // exclude from later picks
        }
        __syncthreads();
    }
}

// ---------------------------------------------------------------------------
// Host launcher
// ---------------------------------------------------------------------------
extern "C" void kernel_launch(void* const* d_in, const int* in_sizes, int n_in,
                              void* d_out, int out_size, void* d_ws, size_t ws_size,
                              hipStream_t stream) {
    (void)in_sizes; (void)n_in; (void)out_size; (void)ws_size;
    const float* enc     = (const float*)d_in[0];   // (B, A, 4)
    const float* logits  = (const float*)d_in[1];   // (B, A, C)
    const float* anchors = (const float*)d_in[2];   // (A, 4)
    float* out = (float*)d_out;                     // (B, 100, 6)

    float* ws     = (float*)d_ws;
    float* dec    = ws;                                           // B*A*4
    float* nmsBox = dec    + (size_t)B_N * A_N * 4;               // B*CC*100*4
    float* nmsSc  = nmsBox + (size_t)B_N * CC_N * MAXDET * 4;     // B*CC*100

    decode_kernel<<<(B_N * A_N + 255) / 256, 256, 0, stream>>>(enc, anchors, dec);
    topk_nms_kernel<<<B_N * CC_N, 256, A_N * sizeof(float), stream>>>(logits, dec, nmsBox, nmsSc);
    final_topk_kernel<<<B_N, 256, 0, stream>>>(nmsBox, nmsSc, out);
}